// H2OTMixSTE_13907104105016
// MI455X (gfx1250) — compile-verified
//
#include <hip/hip_runtime.h>
#include <hip/hip_bf16.h>
#include <cstddef>

// ---------------------------------------------------------------------------
// CDNA5 fp32 WMMA: D(16x16) = A(16x4) x B(4x16) + C, wave32.
// A layout : lanes 0-15 hold rows M=0..15 with {K=k0,k0+1}; lanes 16-31 {K=k0+2,k0+3}
// B layout : lanes 0-15 hold cols N=0..15 with {K=k0,k0+1}; lanes 16-31 {K=k0+2,k0+3}
// C layout : VGPR r, lanes 0-15 -> (M=r, N=lane); lanes 16-31 -> (M=r+8, N=lane-16)
// ---------------------------------------------------------------------------
typedef __attribute__((ext_vector_type(2))) float v2f;
typedef __attribute__((ext_vector_type(8))) float v8f;

static __device__ __forceinline__ v8f wmma_f32_4(v2f a, v2f b, v8f c) {
  return __builtin_amdgcn_wmma_f32_16x16x4_f32(false, a, false, b, (short)0, c,
                                               false, false);
}

// ---- DPP16 cross-lane helpers (stay within 16-lane rows; no LDS traffic) ----
template <int CTRL>
static __device__ __forceinline__ float dppf(float v) {
  return __builtin_bit_cast(
      float, __builtin_amdgcn_update_dpp(0, __builtin_bit_cast(int, v), CTRL,
                                         0xf, 0xf, true));
}
// full reduction within each 16-lane group (halves stay independent)
static __device__ __forceinline__ float rowmax16(float v) {
  v = fmaxf(v, dppf<0xB1>(v));   // quad_perm [1,0,3,2]  (xor 1)
  v = fmaxf(v, dppf<0x4E>(v));   // quad_perm [2,3,0,1]  (xor 2)
  v = fmaxf(v, dppf<0x141>(v));  // row_half_mirror      (xor 4 equiv)
  v = fmaxf(v, dppf<0x140>(v));  // row_mirror           (xor 8 equiv)
  return v;
}
static __device__ __forceinline__ float rowsum16(float v) {
  v += dppf<0xB1>(v);
  v += dppf<0x4E>(v);
  v += dppf<0x141>(v);
  v += dppf<0x140>(v);
  return v;
}
static __device__ __forceinline__ float rowmin16(float v) {
  v = fminf(v, dppf<0xB1>(v));
  v = fminf(v, dppf<0x4E>(v));
  v = fminf(v, dppf<0x141>(v));
  v = fminf(v, dppf<0x140>(v));
  return v;
}
// full 32-lane sum: DPP within rows, one cross-half bpermute at the end
static __device__ __forceinline__ float wavesum32(float v) {
  v = rowsum16(v);
  v += __shfl_xor(v, 16, 32);
  return v;
}

// ---------------------------------------------------------------------------
// Fused GEMM: Y[M,N] = act(A[M,K] @ W[K,N] + bias) (+ resid)
// One wave per 16x32 tile (two WMMA chains share the A fragment);
// 4 waves (128 thr) per block cover 128 cols.
// ---------------------------------------------------------------------------
template <int ACT, bool RESID>
__global__ __launch_bounds__(128) void gemm_kernel(
    const float* __restrict__ A, const float* __restrict__ W,
    const float* __restrict__ bias, const float* __restrict__ resid,
    float* __restrict__ Y, int M, int N, int K) {
  const int lane = threadIdx.x & 31;
  const int wave = threadIdx.x >> 5;
  const int hi = lane >> 4;
  const int l16 = lane & 15;
  const int tile_m = blockIdx.y * 16;
  const int tile_n = (blockIdx.x * 8 + wave * 2) * 16;
  if (tile_n >= N || tile_m >= M) return;  // wave-uniform; EXEC stays full
  const int row = tile_m + l16;
  const int col0 = tile_n + l16;
  const int col1 = col0 + 16;
  const float* Arow = A + (size_t)row * K;

  v8f acc0 = {}, acc1 = {};
  for (int k = 0; k < K; k += 4) {
    __builtin_prefetch(Arow + k + 64, 0, 1);
    v2f a = *(const v2f*)(Arow + k + 2 * hi);
    v2f b0, b1;
    const size_t wr0 = (size_t)(k + 2 * hi) * N;
    const size_t wr1 = (size_t)(k + 1 + 2 * hi) * N;
    b0.x = W[wr0 + col0];
    b0.y = W[wr1 + col0];
    b1.x = W[wr0 + col1];
    b1.y = W[wr1 + col1];
    acc0 = wmma_f32_4(a, b0, acc0);
    acc1 = wmma_f32_4(a, b1, acc1);
  }
  const float bv0 = bias ? bias[col0] : 0.0f;
  const float bv1 = bias ? bias[col1] : 0.0f;
#pragma unroll
  for (int r = 0; r < 8; ++r) {
    const int orow = tile_m + r + 8 * hi;
    float v0 = acc0[r] + bv0;
    float v1 = acc1[r] + bv1;
    if (ACT == 1) {
      v0 = 0.5f * v0 * (1.0f + erff(v0 * 0.70710678118654752f));
      v1 = 0.5f * v1 * (1.0f + erff(v1 * 0.70710678118654752f));
    }
    const size_t o0i = (size_t)orow * N + col0;
    const size_t o1i = (size_t)orow * N + col1;
    if (RESID) {
      v0 += resid[o0i];
      v1 += resid[o1i];
    }
    Y[o0i] = v0;
    Y[o1i] = v1;
  }
}

// ---------------------------------------------------------------------------
// LayerNorm over C=256, one wave per row.
// ---------------------------------------------------------------------------
__global__ __launch_bounds__(256) void layernorm_kernel(
    const float* __restrict__ x, const float* __restrict__ g,
    const float* __restrict__ bta, float* __restrict__ y, int rows) {
  const int wave = threadIdx.x >> 5, lane = threadIdx.x & 31;
  const int row = blockIdx.x * 8 + wave;
  if (row >= rows) return;
  const float* xr = x + (size_t)row * 256;
  float v[8], s = 0.f, s2 = 0.f;
#pragma unroll
  for (int t = 0; t < 8; ++t) {
    float u = xr[t * 32 + lane];
    v[t] = u;
    s += u;
    s2 += u * u;
  }
  s = wavesum32(s);
  s2 = wavesum32(s2);
  const float mean = s * (1.0f / 256.0f);
  const float var = s2 * (1.0f / 256.0f) - mean * mean;
  const float rstd = rsqrtf(var + 1e-5f);
  float* yr = y + (size_t)row * 256;
#pragma unroll
  for (int t = 0; t < 8; ++t) {
    const int c = t * 32 + lane;
    yr[c] = (v[t] - mean) * rstd * g[c] + bta[c];
  }
}

// Row squared-norms over C=256, one wave per row.
__global__ __launch_bounds__(256) void rownorm_kernel(const float* __restrict__ x,
                                                      float* __restrict__ sq,
                                                      int rows) {
  const int wave = threadIdx.x >> 5, lane = threadIdx.x & 31;
  const int row = blockIdx.x * 8 + wave;
  if (row >= rows) return;
  const float* xr = x + (size_t)row * 256;
  float s2 = 0.f;
#pragma unroll
  for (int t = 0; t < 8; ++t) {
    float u = xr[t * 32 + lane];
    s2 += u * u;
  }
  s2 = wavesum32(s2);
  if (lane == 0) sq[row] = s2;
}

// ---------------------------------------------------------------------------
// Flash attention (wave32, d=32): 16 queries per wave, streaming softmax.
// Row statistics live per-lane in the WMMA C-layout; all softmax reductions
// are DPP16-only (no LDS in the reduction path). P bounces through 1KB LDS
// once per chunk for the C->A fragment relayout.
// Grid (Nq/16, H, B). qs/ks = row strides of Q and K/V, os = output stride.
// ---------------------------------------------------------------------------
__global__ __launch_bounds__(32) void flash_kernel(
    const float* __restrict__ Q, const float* __restrict__ K,
    const float* __restrict__ V, float* __restrict__ O, int Nq, int Nk, int qs,
    int ks, int os, float scale) {
  __shared__ float pl[256];  // 16x16 P tile
  const int b = blockIdx.z, h = blockIdx.y;
  const float* Qb = Q + (size_t)b * Nq * qs + h * 32;
  const float* Kb = K + (size_t)b * Nk * ks + h * 32;
  const float* Vb = V + (size_t)b * Nk * ks + h * 32;
  float* Ob = O + (size_t)b * Nq * os + h * 32;
  const int lane = threadIdx.x & 31, hi = lane >> 4, l16 = lane & 15;
  const int qrow = blockIdx.x * 16 + l16;

  float qf[16];
#pragma unroll
  for (int j = 0; j < 8; ++j) {
    qf[2 * j] = Qb[(size_t)qrow * qs + 4 * j + 2 * hi];
    qf[2 * j + 1] = Qb[(size_t)qrow * qs + 4 * j + 1 + 2 * hi];
  }
  float mrow[8], lrow[8];
  v8f o0 = {}, o1 = {};
#pragma unroll
  for (int r = 0; r < 8; ++r) {
    mrow[r] = -1e30f;
    lrow[r] = 0.f;
  }

  for (int kb = 0; kb < Nk; kb += 16) {
    const int key = kb + l16;
    float kf[16];
#pragma unroll
    for (int j = 0; j < 8; ++j) {
      kf[2 * j] = Kb[(size_t)key * ks + 4 * j + 2 * hi];
      kf[2 * j + 1] = Kb[(size_t)key * ks + 4 * j + 1 + 2 * hi];
    }
    v8f s = {};
#pragma unroll
    for (int j = 0; j < 8; ++j) {
      v2f a = {qf[2 * j], qf[2 * j + 1]};
      v2f bb = {kf[2 * j], kf[2 * j + 1]};
      s = wmma_f32_4(a, bb, s);  // S = Q . K^T
    }
#pragma unroll
    for (int r = 0; r < 8; ++r) {
      const float sv = s[r] * scale;
      const float rmax = rowmax16(sv);
      const float mn = fmaxf(mrow[r], rmax);
      const float alpha = expf(mrow[r] - mn);
      const float p = expf(sv - mn);
      const float rs = rowsum16(p);
      lrow[r] = lrow[r] * alpha + rs;
      o0[r] *= alpha;
      o1[r] *= alpha;
      mrow[r] = mn;
      pl[(r + 8 * hi) * 16 + l16] = p;
    }
    __syncthreads();
#pragma unroll
    for (int j = 0; j < 4; ++j) {
      v2f a;
      a.x = pl[l16 * 16 + 4 * j + 2 * hi];
      a.y = pl[l16 * 16 + 4 * j + 1 + 2 * hi];
      const int k0 = kb + 4 * j + 2 * hi;
      v2f b0, b1;
      b0.x = Vb[(size_t)k0 * ks + l16];
      b0.y = Vb[(size_t)(k0 + 1) * ks + l16];
      b1.x = Vb[(size_t)k0 * ks + l16 + 16];
      b1.y = Vb[(size_t)(k0 + 1) * ks + l16 + 16];
      o0 = wmma_f32_4(a, b0, o0);  // O += P . V
      o1 = wmma_f32_4(a, b1, o1);
    }
    __syncthreads();
  }
#pragma unroll
  for (int r = 0; r < 8; ++r) {
    const float inv = 1.0f / lrow[r];
    const int orow = blockIdx.x * 16 + r + 8 * hi;
    Ob[(size_t)orow * os + l16] = o0[r] * inv;
    Ob[(size_t)orow * os + l16 + 16] = o1[r] * inv;
  }
}

// ---------------------------------------------------------------------------
// Fused Gram->distance: dist[b,i,j] = sqrt(max(si+sj-2*x_i.x_j,0))/16 (NT WMMA)
// Two col-tiles per wave; both B fragments are contiguous b64 loads.
// Also atomic-max of dist into dmax[b] (int-bit trick, all values >= 0).
// ---------------------------------------------------------------------------
__global__ __launch_bounds__(128) void dist_kernel(
    const float* __restrict__ X, const float* __restrict__ sq,
    float* __restrict__ dist, unsigned* __restrict__ dmax, int N, int K) {
  const int b = blockIdx.z;
  const float* Xb = X + (size_t)b * N * K;
  float* Db = dist + (size_t)b * N * N;
  const int lane = threadIdx.x & 31;
  const int wave = threadIdx.x >> 5;
  const int hi = lane >> 4;
  const int l16 = lane & 15;
  const int tile_m = blockIdx.y * 16;
  const int tile_n = (blockIdx.x * 8 + wave * 2) * 16;
  const int row = tile_m + l16;
  const int col0 = tile_n + l16;
  const int col1 = col0 + 16;

  v8f acc0 = {}, acc1 = {};
  for (int k = 0; k < K; k += 4) {
    v2f a = *(const v2f*)(Xb + (size_t)row * K + k + 2 * hi);
    v2f b0 = *(const v2f*)(Xb + (size_t)col0 * K + k + 2 * hi);
    v2f b1 = *(const v2f*)(Xb + (size_t)col1 * K + k + 2 * hi);
    acc0 = wmma_f32_4(a, b0, acc0);
    acc1 = wmma_f32_4(a, b1, acc1);
  }
  const float sc0 = sq[b * N + col0];
  const float sc1 = sq[b * N + col1];
  float lmax = 0.f;
#pragma unroll
  for (int r = 0; r < 8; ++r) {
    const int orow = tile_m + r + 8 * hi;
    const float sr = sq[b * N + orow];
    const float d0 = sqrtf(fmaxf(sr + sc0 - 2.0f * acc0[r], 0.0f)) * 0.0625f;
    const float d1 = sqrtf(fmaxf(sr + sc1 - 2.0f * acc1[r], 0.0f)) * 0.0625f;
    Db[(size_t)orow * N + col0] = d0;
    Db[(size_t)orow * N + col1] = d1;
    lmax = fmaxf(lmax, fmaxf(d0, d1));
  }
  atomicMax(&dmax[b], __float_as_uint(lmax));
}

__global__ void init_dmax_kernel(unsigned* dmax) {
  if (threadIdx.x < 4) dmax[threadIdx.x] = 0u;
}

// ---------------------------------------------------------------------------
// 5-NN density: density[b,i] = exp(-mean(d_k^2)) + hash-noise*1e-6
// ---------------------------------------------------------------------------
__global__ __launch_bounds__(32) void knn_density_kernel(
    const float* __restrict__ dist, float* __restrict__ density, int N) {
  const int b = blockIdx.y, i = blockIdx.x, lane = threadIdx.x;
  const float* row = dist + ((size_t)b * N + i) * N;
  float best[5] = {1e30f, 1e30f, 1e30f, 1e30f, 1e30f};
  for (int j = lane; j < N; j += 32) {
    const float v = row[j];
    if (v < best[4]) {
      best[4] = v;
#pragma unroll
      for (int t = 4; t > 0; --t)
        if (best[t] < best[t - 1]) {
          float tmp = best[t];
          best[t] = best[t - 1];
          best[t - 1] = tmp;
        }
    }
  }
  int ptr = 0;
  float sum = 0.f;
#pragma unroll
  for (int r = 0; r < 5; ++r) {
    const float cand = (ptr < 5) ? best[ptr] : 1e30f;
    unsigned long long red =
        (((unsigned long long)__float_as_uint(cand)) << 6) | (unsigned)lane;
    for (int m = 16; m; m >>= 1) {
      unsigned lo = (unsigned)red, hw = (unsigned)(red >> 32);
      lo = __shfl_xor(lo, m, 32);
      hw = __shfl_xor(hw, m, 32);
      const unsigned long long other = ((unsigned long long)hw << 32) | lo;
      red = (other < red) ? other : red;
    }
    const float v = __uint_as_float((unsigned)(red >> 6));
    sum += v * v;
    if ((int)(red & 63ull) == lane) ptr++;
  }
  if (lane == 0) {
    float d = expf(-(sum * 0.2f));
    unsigned h = (unsigned)i * 2654435761u ^ ((unsigned)b * 2246822519u) ^
                 0x85EBCA6Bu;
    h ^= h >> 16;
    h *= 0x7FEB352Du;
    h ^= h >> 15;
    d += (float)(h >> 8) * (1.0f / 16777216.0f) * 1e-6f;
    density[(size_t)b * N + i] = d;
  }
}

// score[b,i] = density[i] * min_{j: density[j]>density[i]} dist[i,j]  (else dmax)
__global__ __launch_bounds__(32) void score_kernel(
    const float* __restrict__ dist, const float* __restrict__ density,
    const unsigned* __restrict__ dmax, float* __restrict__ score, int N) {
  const int b = blockIdx.y, i = blockIdx.x, lane = threadIdx.x;
  const float* row = dist + ((size_t)b * N + i) * N;
  const float* db = density + (size_t)b * N;
  const float di = db[i];
  float m = 1e30f;
  for (int j = lane; j < N; j += 32)
    if (db[j] > di) m = fminf(m, row[j]);
  m = rowmin16(m);
  m = fminf(m, __shfl_xor(m, 16, 32));
  if (lane == 0) {
    if (m >= 1e29f) m = __uint_as_float(dmax[b]);
    score[(size_t)b * N + i] = m * di;
  }
}

// Bitonic top-512 (descending); order of selected indices is irrelevant
// downstream (attention over keys is permutation-invariant).
__global__ __launch_bounds__(1024) void topk_kernel(
    const float* __restrict__ score, int* __restrict__ idx_out, int N,
    int Ksel) {
  __shared__ float sk[2048];
  __shared__ int sv[2048];
  const int b = blockIdx.x, tid = threadIdx.x;
  for (int i = tid; i < N; i += 1024) {
    sk[i] = score[(size_t)b * N + i];
    sv[i] = i;
  }
  __syncthreads();
  for (int k = 2; k <= N; k <<= 1)
    for (int j = k >> 1; j > 0; j >>= 1) {
      for (int i = tid; i < N; i += 1024) {
        const int ixj = i ^ j;
        if (ixj > i) {
          const bool desc = ((i & k) == 0);
          const float a = sk[i], c = sk[ixj];
          const bool sw = desc ? (a < c) : (a > c);
          if (sw) {
            sk[i] = c;
            sk[ixj] = a;
            const int t = sv[i];
            sv[i] = sv[ixj];
            sv[ixj] = t;
          }
        }
      }
      __syncthreads();
    }
  for (int i = tid; i < Ksel; i += 1024) idx_out[b * Ksel + i] = sv[i];
}

__global__ __launch_bounds__(64) void gather_kernel(
    const float* __restrict__ x, const int* __restrict__ idx,
    float* __restrict__ xd, int N, int Ksel) {
  const int b = blockIdx.y, j = blockIdx.x;
  const int src = idx[b * Ksel + j];
  const float* s = x + ((size_t)b * N + src) * 256;
  float* d = xd + ((size_t)b * Ksel + j) * 256;
  for (int t = threadIdx.x; t < 256; t += 64) d[t] = s[t];
}

// ---------------------------------------------------------------------------
// Host-side orchestration
// ---------------------------------------------------------------------------
namespace {
constexpr int Bn = 4, Fn = 2048, CN = 512;

struct BlockP {
  const float *ln1_g, *ln1_b, *qkv_w, *proj_w, *proj_b, *ln2_g, *ln2_b, *fc1_w,
      *fc1_b, *fc2_w, *fc2_b;
};

BlockP load_block(void* const* d_in, int base) {
  BlockP p;
  p.ln1_g = (const float*)d_in[base + 0];
  p.ln1_b = (const float*)d_in[base + 1];
  p.qkv_w = (const float*)d_in[base + 2];
  p.proj_w = (const float*)d_in[base + 3];
  p.proj_b = (const float*)d_in[base + 4];
  p.ln2_g = (const float*)d_in[base + 5];
  p.ln2_b = (const float*)d_in[base + 6];
  p.fc1_w = (const float*)d_in[base + 7];
  p.fc1_b = (const float*)d_in[base + 8];
  p.fc2_w = (const float*)d_in[base + 9];
  p.fc2_b = (const float*)d_in[base + 10];
  return p;
}

void run_block(const BlockP& p, float* x, int Ntok, float* xn, float* qkv,
               float* attno, float* hbuf, hipStream_t st) {
  const int Mrows = Bn * Ntok;
  layernorm_kernel<<<Mrows / 8, 256, 0, st>>>(x, p.ln1_g, p.ln1_b, xn, Mrows);
  gemm_kernel<0, false><<<dim3(768 / 128, Mrows / 16), 128, 0, st>>>(
      xn, p.qkv_w, nullptr, nullptr, qkv, Mrows, 768, 256);
  flash_kernel<<<dim3(Ntok / 16, 8, Bn), 32, 0, st>>>(
      qkv, qkv + 256, qkv + 512, attno, Ntok, Ntok, 768, 768, 256,
      0.17677669529663687f);
  gemm_kernel<0, true><<<dim3(256 / 128, Mrows / 16), 128, 0, st>>>(
      attno, p.proj_w, p.proj_b, x, x, Mrows, 256, 256);
  layernorm_kernel<<<Mrows / 8, 256, 0, st>>>(x, p.ln2_g, p.ln2_b, xn, Mrows);
  gemm_kernel<1, false><<<dim3(1024 / 128, Mrows / 16), 128, 0, st>>>(
      xn, p.fc1_w, p.fc1_b, nullptr, hbuf, Mrows, 1024, 256);
  gemm_kernel<0, true><<<dim3(256 / 128, Mrows / 16), 128, 0, st>>>(
      hbuf, p.fc2_w, p.fc2_b, x, x, Mrows, 256, 1024);
}
}  // namespace

extern "C" void kernel_launch(void* const* d_in, const int* in_sizes, int n_in,
                              void* d_out, int out_size, void* d_ws,
                              size_t ws_size, hipStream_t stream) {
  (void)in_sizes;
  (void)n_in;
  (void)out_size;
  (void)ws_size;
  const float* x_in = (const float*)d_in[0];
  const BlockP b1a = load_block(d_in, 1);
  const BlockP b1b = load_block(d_in, 12);
  const BlockP b2a = load_block(d_in, 23);
  const BlockP b2b = load_block(d_in, 34);
  const float* q_w = (const float*)d_in[45];
  const float* k_w = (const float*)d_in[46];
  const float* v_w = (const float*)d_in[47];
  const float* cproj_w = (const float*)d_in[48];
  const float* cproj_b = (const float*)d_in[49];

  float* ws = (float*)d_ws;
  float* xbuf = ws;                   // 2,097,152
  float* xn = xbuf + 2097152;         // 2,097,152
  float* attno = xn + 2097152;        // 2,097,152
  float* qkv = attno + 2097152;       // 6,291,456
  float* hbuf = qkv + 6291456;        // 8,388,608
  float* dist = hbuf + 8388608;       // 16,777,216
  float* sq = dist + 16777216;        // 8192
  float* density = sq + 8192;         // 8192
  float* score = density + 8192;      // 8192
  unsigned* dmax = (unsigned*)(score + 8192);  // 64
  int* idx = (int*)(dmax + 64);                // 2048
  float* xd = (float*)(idx + 2048);            // 524,288

  hipMemcpyAsync(xbuf, x_in, sizeof(float) * 2097152,
                 hipMemcpyDeviceToDevice, stream);

  // ---- blocks1 on (B,2048,256)
  run_block(b1a, xbuf, Fn, xn, qkv, attno, hbuf, stream);
  run_block(b1b, xbuf, Fn, xn, qkv, attno, hbuf, stream);

  // ---- DPC-KNN clustering -> index_down (top-512 per batch)
  rownorm_kernel<<<8192 / 8, 256, 0, stream>>>(xbuf, sq, 8192);
  init_dmax_kernel<<<1, 32, 0, stream>>>(dmax);
  dist_kernel<<<dim3(2048 / 128, 2048 / 16, Bn), 128, 0, stream>>>(
      xbuf, sq, dist, dmax, 2048, 256);
  knn_density_kernel<<<dim3(2048, Bn), 32, 0, stream>>>(dist, density, 2048);
  score_kernel<<<dim3(2048, Bn), 32, 0, stream>>>(dist, density, dmax, score,
                                                  2048);
  topk_kernel<<<Bn, 1024, 0, stream>>>(score, idx, 2048, 512);
  gather_kernel<<<dim3(512, Bn), 64, 0, stream>>>(xbuf, idx, xd, 2048, 512);

  // ---- blocks2 on (B,512,256)
  run_block(b2a, xd, CN, xn, qkv, attno, hbuf, stream);
  run_block(b2b, xd, CN, xn, qkv, attno, hbuf, stream);

  // ---- cross attention: queries = x (2048), keys/values = x_down (512)
  float* qb = hbuf;             // 2,097,152
  float* kb = hbuf + 2097152;   // 524,288
  float* vb = kb + 524288;      // 524,288
  gemm_kernel<0, false><<<dim3(256 / 128, 8192 / 16), 128, 0, stream>>>(
      xbuf, q_w, nullptr, nullptr, qb, 8192, 256, 256);
  gemm_kernel<0, false><<<dim3(256 / 128, 2048 / 16), 128, 0, stream>>>(
      xd, k_w, nullptr, nullptr, kb, 2048, 256, 256);
  gemm_kernel<0, false><<<dim3(256 / 128, 2048 / 16), 128, 0, stream>>>(
      xd, v_w, nullptr, nullptr, vb, 2048, 256, 256);
  flash_kernel<<<dim3(2048 / 16, 8, Bn), 32, 0, stream>>>(
      qb, kb, vb, attno, 2048, 512, 256, 256, 256, 0.17677669529663687f);
  gemm_kernel<0, true><<<dim3(256 / 128, 8192 / 16), 128, 0, stream>>>(
      attno, cproj_w, cproj_b, xbuf, (float*)d_out, 8192, 256, 256);
}